// LongTermModel_85126251806847
// MI455X (gfx1250) — compile-verified
//
#include <hip/hip_runtime.h>
#include <hip/hip_bf16.h>
#include <stdint.h>

// Segment-mean embedding reduction for MI455X (gfx1250, wave32).
// Data path: global -> (global_load_async_to_lds_b128, ASYNCcnt) -> LDS ->
// ds_load_b128 -> VALU accumulate -> non-temporal b128 stores.

typedef float v4f __attribute__((ext_vector_type(4)));

#define EMB 128
#define NUM_DAYS 100000
#define DPW 8              // days per wave (amortizes binary search 8x)
#define WAVES_PER_BLOCK 4  // 128 threads/block -> 64KB static LDS
#define CHUNK 8            // rows per async double-buffer chunk

__device__ __forceinline__ int lower_bound_i32(const int* __restrict__ a, int n, int key) {
    int lo = 0, hi = n;
    #pragma unroll 1
    while (lo < hi) {
        int mid = (lo + hi) >> 1;
        if (a[mid] < key) lo = mid + 1; else hi = mid;
    }
    return lo;
}

// Issue one 8-row chunk: per row, one async b128 per table; 32 lanes x 16B = 512B row.
__device__ __forceinline__ void issue_chunk(const float* __restrict__ news,
                                            const float* __restrict__ cat,
                                            int nid, int cid,
                                            uint32_t baseN, uint32_t baseC,
                                            int parity, int lane) {
    const uint64_t laneByte = (uint64_t)(uint32_t)lane * 16u;
    #pragma unroll
    for (int j = 0; j < CHUNK; ++j) {
        int idn = __shfl(nid, j, 32);
        int idc = __shfl(cid, j, 32);
        uint64_t an = (uint64_t)(uintptr_t)news + (uint64_t)(uint32_t)idn * (EMB * 4u) + laneByte;
        uint64_t ac = (uint64_t)(uintptr_t)cat  + (uint64_t)(uint32_t)idc * (EMB * 4u) + laneByte;
        uint32_t ln = baseN + (uint32_t)((parity * CHUNK + j) * (EMB * 4)) + (uint32_t)lane * 16u;
        uint32_t lc = baseC + (uint32_t)((parity * CHUNK + j) * (EMB * 4)) + (uint32_t)lane * 16u;
        asm volatile("global_load_async_to_lds_b128 %0, %1, off" :: "v"(ln), "v"(an) : "memory");
        asm volatile("global_load_async_to_lds_b128 %0, %1, off" :: "v"(lc), "v"(ac) : "memory");
    }
}

__device__ __forceinline__ void flush_day(float* __restrict__ Z, int day,
                                          v4f acc, int cnt, int lane) {
    float s = 1.0f / (float)((cnt > 0) ? cnt : 1);
    v4f v = acc * s;
    // Z is write-once streaming output: non-temporal, keep L2 for the tables.
    __builtin_nontemporal_store(v, (v4f*)(Z + (size_t)day * EMB + (size_t)lane * 4));
}

__global__ __launch_bounds__(WAVES_PER_BLOCK * 32)
void LongTermModel_segmean_kernel(const float* __restrict__ news,
                                  const float* __restrict__ cat,
                                  const int* __restrict__ nids,
                                  const int* __restrict__ cids,
                                  const int* __restrict__ segs,
                                  const float* __restrict__ gaps,
                                  float* __restrict__ Z,
                                  float* __restrict__ dt,
                                  int total) {
    __shared__ __align__(16) float s_n[WAVES_PER_BLOCK][2][CHUNK][EMB];
    __shared__ __align__(16) float s_c[WAVES_PER_BLOCK][2][CHUNK][EMB];

    const int lane  = threadIdx.x & 31;
    const int wslot = threadIdx.x >> 5;
    const int gw    = blockIdx.x * WAVES_PER_BLOCK + wslot;
    const int d0    = gw * DPW;
    if (d0 >= NUM_DAYS) return;

    // delta_t passthrough
    if (lane < DPW && d0 + lane < NUM_DAYS) dt[d0 + lane] = gaps[d0 + lane];

    const int start = lower_bound_i32(segs, total, d0);
    const int end   = lower_bound_i32(segs, total, d0 + DPW);

    const uint32_t baseN = (uint32_t)(uintptr_t)&s_n[wslot][0][0][0];
    const uint32_t baseC = (uint32_t)(uintptr_t)&s_c[wslot][0][0][0];

    int cur = d0;
    v4f acc = {0.f, 0.f, 0.f, 0.f};
    int cnt = 0;

    if (start < end) {
        const int rowlim = end - 1;
        // Prologue: ids + async issue for chunk 0.
        int rj  = min(start + (lane & 7), rowlim);
        int nid = nids[rj];
        int cid = cids[rj];
        int sgv = segs[rj];
        issue_chunk(news, cat, nid, cid, baseN, baseC, 0, lane);

        int r = start;
        int parity = 0;
        #pragma unroll 1
        while (true) {
            const int nsub = min(CHUNK, end - r);
            const int rn   = r + CHUNK;
            const bool more = rn < end;
            int nid2 = 0, cid2 = 0, sg2 = 0;
            if (more) {
                int rj2 = min(rn + (lane & 7), rowlim);
                nid2 = nids[rj2];
                cid2 = cids[rj2];
                sg2  = segs[rj2];
                issue_chunk(news, cat, nid2, cid2, baseN, baseC, parity ^ 1, lane);
                // 16 ops of the next chunk in flight; ASYNCcnt completes in order,
                // so <=16 outstanding means the current chunk has fully landed in LDS.
                asm volatile("s_wait_asynccnt 16" ::: "memory");
            } else {
                asm volatile("s_wait_asynccnt 0" ::: "memory");
            }

            #pragma unroll 1
            for (int j = 0; j < nsub; ++j) {
                int sg = __shfl(sgv, j, 32);
                if (sg != cur) {
                    flush_day(Z, cur, acc, cnt, lane);
                    for (int dd = cur + 1; dd < sg; ++dd) {
                        v4f z = {0.f, 0.f, 0.f, 0.f};
                        flush_day(Z, dd, z, 0, lane);
                    }
                    cur = sg;
                    acc = (v4f){0.f, 0.f, 0.f, 0.f};
                    cnt = 0;
                }
                const v4f n = *(const v4f*)&s_n[wslot][parity][j][lane * 4];
                const v4f c = *(const v4f*)&s_c[wslot][parity][j][lane * 4];
                acc += n + c;
                ++cnt;
            }

            if (!more) break;
            nid = nid2; cid = cid2; sgv = sg2;
            r = rn;
            parity ^= 1;
        }
    }

    // Epilogue: flush last open day, zero-fill remaining (possibly empty) days.
    flush_day(Z, cur, acc, cnt, lane);
    for (int dd = cur + 1; dd < d0 + DPW; ++dd) {
        v4f z = {0.f, 0.f, 0.f, 0.f};
        flush_day(Z, dd, z, 0, lane);
    }
}

extern "C" void kernel_launch(void* const* d_in, const int* in_sizes, int n_in,
                              void* d_out, int out_size, void* d_ws, size_t ws_size,
                              hipStream_t stream) {
    const float* news = (const float*)d_in[0];   // [NUM_NEWS, EMB] f32
    const float* cat  = (const float*)d_in[1];   // [NUM_CAT, EMB]  f32
    const int*   nids = (const int*)d_in[2];     // [TOTAL]
    const int*   cids = (const int*)d_in[3];     // [TOTAL]
    const int*   segs = (const int*)d_in[4];     // [TOTAL], sorted
    const float* gaps = (const float*)d_in[5];   // [NUM_DAYS]
    const int total = in_sizes[2];

    float* Z  = (float*)d_out;                       // [NUM_DAYS, EMB]
    float* dt = Z + (size_t)NUM_DAYS * EMB;          // [NUM_DAYS]

    const int waves  = (NUM_DAYS + DPW - 1) / DPW;                       // 12500
    const int blocks = (waves + WAVES_PER_BLOCK - 1) / WAVES_PER_BLOCK;  // 3125

    hipLaunchKernelGGL(LongTermModel_segmean_kernel,
                       dim3(blocks), dim3(WAVES_PER_BLOCK * 32), 0, stream,
                       news, cat, nids, cids, segs, gaps, Z, dt, total);
}